// WeightedLoss_58703613002104
// MI455X (gfx1250) — compile-verified
//
#include <hip/hip_runtime.h>

// ---------------------------------------------------------------------------
// WeightedLoss: mean(where(target==1, 1 - sigmoid(pred), 0.1)) over 64M elems.
// Bandwidth-bound: 8B/elem read, 512MB total @ 23.3 TB/s -> ~22us floor.
//   - f32x4 / i32x4 nontemporal b128 loads, 32-bit indexing (GVS addressing)
//   - 1 - sigmoid(p) = 0.5 - 0.5*tanh(p/2): single v_tanh_f32 trans op
//   - wave reduction via V_WMMA_F32_16X16X4_F32 (A = lane partials, B = ones)
//   - two-pass deterministic tree: block partials in d_ws, one-wave finalize
// ---------------------------------------------------------------------------

typedef __attribute__((ext_vector_type(4))) float f32x4;
typedef __attribute__((ext_vector_type(4))) int   i32x4;
typedef __attribute__((ext_vector_type(2))) float v2f;
typedef __attribute__((ext_vector_type(8))) float v8f;

#define NEG_WEIGHT 0.1f
#define THREADS    256        // 8 wave32 per block
#define NWAVES     (THREADS / 32)

// Reduce one float per lane across the 32-lane wave using the f32 WMMA.
// A (16x4 f32): lanes 0-15 hold {A[L][0], A[L][1]} = {partial, 0};
//               lanes 16-31 hold {A[L-16][2], A[L-16][3]} = {partial, 0}.
// B (4x16) = all ones. D[m][n] = p_m + p_{m+16} (replicated over n).
// D layout: VGPR r holds row r (lanes 0-15) / row r+8 (lanes 16-31), so the
// 8-VGPR sum gives rows 0..7 / 8..15; xor-16 add completes the 32-lane sum.
__device__ __forceinline__ float wave_reduce_wmma(float partial) {
    v2f a; a.x = partial; a.y = 0.0f;
    v2f b; b.x = 1.0f;    b.y = 1.0f;
    v8f c = {};
    v8f d = __builtin_amdgcn_wmma_f32_16x16x4_f32(
        /*neg_a=*/false, a, /*neg_b=*/false, b,
        /*c_mod=*/(short)0, c, /*reuse_a=*/false, /*reuse_b=*/false);
    float s = ((d[0] + d[1]) + (d[2] + d[3])) + ((d[4] + d[5]) + (d[6] + d[7]));
    s += __shfl_xor(s, 16, 32);
    return s;
}

__device__ __forceinline__ float elem_loss(float p, int t) {
#if __has_builtin(__builtin_amdgcn_tanhf)
    // 1 - sigmoid(p) = 0.5 - 0.5*tanh(p/2): one v_tanh_f32 + mul + fma
    float sn = __builtin_fmaf(-0.5f, __builtin_amdgcn_tanhf(0.5f * p), 0.5f);
#else
    // 1 - sigmoid(p) = 1/(1+exp(p)): v_exp_f32 + v_rcp_f32
    float sn = __builtin_amdgcn_rcpf(1.0f + __expf(p));
#endif
    return (t == 1) ? sn : NEG_WEIGHT;             // v_cndmask
}

__global__ void __launch_bounds__(THREADS)
loss_partial_kernel(const float* __restrict__ pred,
                    const int*   __restrict__ targ,
                    float*       __restrict__ partials,
                    int n) {
    const unsigned nvec  = (unsigned)n >> 2;               // # of 4-elem vectors
    const unsigned gsize = gridDim.x * THREADS;            // grid stride (vectors)
    const unsigned gtid  = blockIdx.x * THREADS + threadIdx.x;

    const f32x4* p4 = (const f32x4*)pred;
    const i32x4* t4 = (const i32x4*)targ;

    float acc = 0.0f;
    unsigned i = gtid;

    // Unroll-by-4 grid-stride: 8 nontemporal b128 loads in flight per step.
    for (; i + 3u * gsize < nvec; i += 4u * gsize) {
        f32x4 p0 = __builtin_nontemporal_load(p4 + i);
        f32x4 p1 = __builtin_nontemporal_load(p4 + i + gsize);
        f32x4 p2 = __builtin_nontemporal_load(p4 + i + 2u * gsize);
        f32x4 p3 = __builtin_nontemporal_load(p4 + i + 3u * gsize);
        i32x4 t0 = __builtin_nontemporal_load(t4 + i);
        i32x4 t1 = __builtin_nontemporal_load(t4 + i + gsize);
        i32x4 t2 = __builtin_nontemporal_load(t4 + i + 2u * gsize);
        i32x4 t3 = __builtin_nontemporal_load(t4 + i + 3u * gsize);
#pragma unroll
        for (int k = 0; k < 4; ++k) {
            acc += elem_loss(p0[k], t0[k]);
            acc += elem_loss(p1[k], t1[k]);
            acc += elem_loss(p2[k], t2[k]);
            acc += elem_loss(p3[k], t3[k]);
        }
    }
    // Vector remainder.
    for (; i < nvec; i += gsize) {
        f32x4 p = __builtin_nontemporal_load(p4 + i);
        i32x4 t = __builtin_nontemporal_load(t4 + i);
#pragma unroll
        for (int k = 0; k < 4; ++k) acc += elem_loss(p[k], t[k]);
    }
    // Scalar tail (n % 4), block 0 only; reconverges before the WMMA.
    if (blockIdx.x == 0) {
        for (unsigned j = (nvec << 2) + threadIdx.x; j < (unsigned)n; j += THREADS)
            acc += elem_loss(pred[j], targ[j]);
    }

    // Wave-level reduction via WMMA (EXEC all-1s: full block, converged here).
    float wsum = wave_reduce_wmma(acc);

    __shared__ float lds[NWAVES];
    const int lane = threadIdx.x & 31;
    const int wave = threadIdx.x >> 5;
    if (lane == 0) lds[wave] = wsum;
    __syncthreads();

    if (threadIdx.x == 0) {
        float b = lds[0];
#pragma unroll
        for (int w = 1; w < NWAVES; ++w) b += lds[w];
        partials[blockIdx.x] = b;
    }
}

__global__ void __launch_bounds__(32)
loss_finalize_kernel(const float* __restrict__ partials, int nparts,
                     float* __restrict__ out, float inv_n) {
    float acc = 0.0f;
    for (int i = threadIdx.x; i < nparts; i += 32) acc += partials[i];
    float s = wave_reduce_wmma(acc);   // single full wave -> EXEC all-1s
    if (threadIdx.x == 0) out[0] = s * inv_n;
}

extern "C" void kernel_launch(void* const* d_in, const int* in_sizes, int n_in,
                              void* d_out, int out_size, void* d_ws, size_t ws_size,
                              hipStream_t stream) {
    const float* pred = (const float*)d_in[0];
    const int*   targ = (const int*)d_in[1];
    const int    n    = in_sizes[0];          // 64*1024*1024 fits in int32

    float* partials = (float*)d_ws;

    int blocks = 1024;                        // 256K threads -> 64 vec4 each
    if ((size_t)blocks * sizeof(float) > ws_size)
        blocks = (int)(ws_size / sizeof(float));
    if (blocks < 1) blocks = 1;

    loss_partial_kernel<<<blocks, THREADS, 0, stream>>>(pred, targ, partials, n);
    loss_finalize_kernel<<<1, 32, 0, stream>>>(partials, blocks, (float*)d_out,
                                               1.0f / (float)n);
}